// PointFeatureAlignment_17506286698530
// MI455X (gfx1250) — compile-verified
//
#include <hip/hip_runtime.h>
#include <hip/hip_bf16.h>

#define H_IMG 224
#define W_IMG 224
#define B_    4
#define C_    256
#define M_    1024
#define N_    16384
#define HW_   (H_IMG * W_IMG)
#define OH_   56
#define OW_   56

typedef __attribute__((ext_vector_type(2))) float v2f;
typedef __attribute__((ext_vector_type(8))) float v8f;

#if defined(__has_builtin)
#if __has_builtin(__builtin_amdgcn_wmma_f32_16x16x4_f32)
#define HAVE_WMMA_F32X4 1
#endif
#endif

// -------------------- transpose xyz_feats (B,C,M) -> (B,M,C) --------------------
__global__ __launch_bounds__(256) void transpose_feats_kernel(
    const float* __restrict__ xf, float* __restrict__ Ft) {
  const int bm = blockIdx.x;            // B*M blocks
  const int b  = bm >> 10;
  const int m  = bm & (M_ - 1);
  const int c  = threadIdx.x;
  Ft[((long)b * M_ + m) * C_ + c] = xf[((long)b * C_ + c) * M_ + m];
}

// -------------------- phase A: WMMA distance top-3 + weights + pixel --------------------
// Block = 256 threads = 8 waves; each wave handles 16 points vs all 1024 centers.
// rec layout per point (8 dwords): [i0,i1,i2](int) [w0,w1,w2](f32) [pix](int) [pad]
__global__ __launch_bounds__(256) void topk_wmma_kernel(
    const float* __restrict__ points,   // (B,N,3)
    const float* __restrict__ centers,  // (B,M,3)
    const float* __restrict__ Kmat,     // 3x3 row-major
    float* __restrict__ rec) {
  // Centers staged as float4 (cx, cy, cz, |c|^2): each lane's A-fragment is one
  // contiguous 8-byte LDS read. Lanes 0-15 take (cx,cy) [K=0,1], lanes 16-31
  // take (cz,|c|^2) [K=2,3].
  __shared__ float4 s_c[M_];            // 16 KB
  const int b = blockIdx.x >> 7;        // 128 blocks per batch (B*N/128 = 512 blocks)

  for (int m = threadIdx.x; m < M_; m += 256) {
    const float cx = centers[((long)b * M_ + m) * 3 + 0];
    const float cy = centers[((long)b * M_ + m) * 3 + 1];
    const float cz = centers[((long)b * M_ + m) * 3 + 2];
    s_c[m] = make_float4(cx, cy, cz, cx * cx + cy * cy + cz * cz);
  }
  __syncthreads();

  const int lane  = threadIdx.x & 31;
  const int wave  = threadIdx.x >> 5;
  const bool hi   = lane >= 16;
  const int  lo16 = lane & 15;
  const int pbase = (blockIdx.x & 127) * 128 + wave * 16;  // point base within batch
  const int n     = pbase + lo16;                          // this lane's point

  // Lanes L and L+16 handle the same point (column L&15 of the D tile).
  const float px = points[((long)b * N_ + n) * 3 + 0];
  const float py = points[((long)b * N_ + n) * 3 + 1];
  const float pz = points[((long)b * N_ + n) * 3 + 2];

  // B-matrix (4x16): column n = (-2px,-2py,-2pz,1); lanes 0-15 hold K=0,1; 16-31 K=2,3
  v2f bfrag;
  bfrag.x = hi ? (-2.0f * pz) : (-2.0f * px);
  bfrag.y = hi ? 1.0f         : (-2.0f * py);

  // Per-lane LDS pointer for the A-fragment (lane-dependent offset hoisted).
  const float* s_base = ((const float*)s_c) + lo16 * 4 + (hi ? 2 : 0);

  float s0 = 3.0e38f, s1 = 3.0e38f, s2 = 3.0e38f;
  int   i0 = 0,       i1 = 0,       i2 = 0;
  const int mofs = hi ? 8 : 0;

#pragma unroll 2
  for (int t = 0; t < M_ / 16; ++t) {
    // A-matrix (16x4): row m = (cx,cy,cz,|c|^2)
    const v2f afrag = *(const v2f*)(s_base + t * 64);

    v8f d;
#ifdef HAVE_WMMA_F32X4
    v8f czero = {};
    // D[i][j] = sum_k A[i][k]*B[k][j] = |c_i|^2 - 2 p_j . c_i
    d = __builtin_amdgcn_wmma_f32_16x16x4_f32(
        /*neg_a=*/false, afrag, /*neg_b=*/false, bfrag,
        /*c_mod=*/(short)0, czero, /*reuse_a=*/false, /*reuse_b=*/false);
#else
    #pragma unroll
    for (int r = 0; r < 8; ++r) {
      const float4 cc = s_c[t * 16 + r + mofs];
      d[r] = cc.w - 2.0f * (px * cc.x + py * cc.y + pz * cc.z);
    }
#endif
    // Wave-uniform fast skip: only run the insert chain if any lane's chunk
    // minimum would enter that lane's current top-3.
    const float m01 = fminf(d[0], d[1]);
    const float m23 = fminf(d[2], d[3]);
    const float m45 = fminf(d[4], d[5]);
    const float m67 = fminf(d[6], d[7]);
    const float dmin = fminf(fminf(m01, m23), fminf(m45, m67));
    if (__any(dmin < s2)) {
      // D layout: VGPR r, lane L: row M = r + 8*(L>=16), col N = L&15.
      // Branchless running top-3 (v_cmp + v_cndmask, no EXEC divergence).
      #pragma unroll
      for (int r = 0; r < 8; ++r) {
        const float s = d[r];
        const int   m = t * 16 + r + mofs;
        const bool c0 = s < s0;
        const bool c1 = s < s1;
        const bool c2 = s < s2;
        const float ns2 = c1 ? s1 : (c2 ? s : s2);
        const int   ni2 = c1 ? i1 : (c2 ? m : i2);
        const float ns1 = c0 ? s0 : (c1 ? s : s1);
        const int   ni1 = c0 ? i0 : (c1 ? m : i1);
        const float ns0 = c0 ? s  : s0;
        const int   ni0 = c0 ? m  : i0;
        s2 = ns2; i2 = ni2; s1 = ns1; i1 = ni1; s0 = ns0; i0 = ni0;
      }
    }
  }

  // Merge sorted triples between lane and lane^16 (same point, disjoint center rows),
  // with (score, index) lexicographic tie-break to match top_k stability.
  const float t0 = __shfl_xor(s0, 16); const int j0 = __shfl_xor(i0, 16);
  const float t1 = __shfl_xor(s1, 16); const int j1 = __shfl_xor(i1, 16);
  const float t2 = __shfl_xor(s2, 16); const int j2 = __shfl_xor(i2, 16);
  #pragma unroll
  for (int k = 0; k < 3; ++k) {
    const float s = (k == 0) ? t0 : (k == 1) ? t1 : t2;
    const int   m = (k == 0) ? j0 : (k == 1) ? j1 : j2;
    const bool c0 = (s < s0) || (s == s0 && m < i0);
    const bool c1 = (s < s1) || (s == s1 && m < i1);
    const bool c2 = (s < s2) || (s == s2 && m < i2);
    const float ns2 = c1 ? s1 : (c2 ? s : s2);
    const int   ni2 = c1 ? i1 : (c2 ? m : i2);
    const float ns1 = c0 ? s0 : (c1 ? s : s1);
    const int   ni1 = c0 ? i0 : (c1 ? m : i1);
    const float ns0 = c0 ? s  : s0;
    const int   ni0 = c0 ? m  : i0;
    s2 = ns2; i2 = ni2; s1 = ns1; i1 = ni1; s0 = ns0; i0 = ni0;
  }

  if (lane < 16) {
    const float psq = px * px + py * py + pz * pz;
    const float d0 = s0 + psq, d1v = s1 + psq, d2v = s2 + psq;
    float w0 = 1.0f / (d0 + 1e-8f);
    float w1 = 1.0f / (d1v + 1e-8f);
    float w2 = 1.0f / (d2v + 1e-8f);
    const float inv = 1.0f / (w0 + w1 + w2);
    w0 *= inv; w1 *= inv; w2 *= inv;

    // Projection: uvz = K @ p, uv = uvz[:2]/max(z,1e-8)
    const float fx = Kmat[0], cxk = Kmat[2], fy = Kmat[4], cyk = Kmat[5];
    const float zz = fmaxf(pz, 1e-8f);
    const float u = floorf((fx * px + cxk * pz) / zz);
    const float v = floorf((fy * py + cyk * pz) / zz);
    const int ui = (int)fminf(fmaxf(u, 0.0f), (float)(W_IMG - 1));
    const int vi = (int)fminf(fmaxf(v, 0.0f), (float)(H_IMG - 1));
    const int pix = ui + vi * W_IMG;

    float* R = rec + ((long)b * N_ + n) * 8;
    int*   RI = (int*)R;
    RI[0] = i0; RI[1] = i1; RI[2] = i2;
    R[3] = w0; R[4] = w1; R[5] = w2;
    RI[6] = pix;
  }
}

// -------------------- phase B: gather/interp + scatter-add --------------------
__global__ __launch_bounds__(256) void scatter_kernel(
    const float* __restrict__ Ft,   // (B,M,C)
    const float* __restrict__ rec,
    float* __restrict__ fmap,       // (B,HW,C)
    float* __restrict__ cnt) {      // (B,HW)
  const long p = blockIdx.x;        // B*N blocks
  const int  b = (int)(p >> 14);
  const float* R = rec + p * 8;
  const int* RI = (const int*)R;
  const int i0 = RI[0], i1 = RI[1], i2 = RI[2], pix = RI[6];
  const float w0 = R[3], w1 = R[4], w2 = R[5];
  const int c = threadIdx.x;
  const float* F = Ft + (long)b * M_ * C_;
  const float val = w0 * F[(long)i0 * C_ + c] +
                    w1 * F[(long)i1 * C_ + c] +
                    w2 * F[(long)i2 * C_ + c];
  atomicAdd(&fmap[((long)b * HW_ + pix) * C_ + c], val);
  if (c == 0) atomicAdd(&cnt[(long)b * HW_ + pix], 1.0f);
}

// -------------------- phase C: normalize by count + 4x4 avg-pool --------------------
__global__ __launch_bounds__(256) void pool_kernel(
    const float* __restrict__ fmap, const float* __restrict__ cnt,
    float* __restrict__ out) {
  const int ob = blockIdx.x;        // B*56*56 blocks
  const int b  = ob / (OH_ * OW_);
  const int r  = ob % (OH_ * OW_);
  const int oh = r / OW_, ow = r % OW_;
  const int c  = threadIdx.x;
  float acc = 0.0f;
  #pragma unroll
  for (int dy = 0; dy < 4; ++dy) {
    #pragma unroll
    for (int dx = 0; dx < 4; ++dx) {
      const int pix = (oh * 4 + dy) * W_IMG + (ow * 4 + dx);
      const float cv = cnt[(long)b * HW_ + pix];
      const float inv = 1.0f / fmaxf(cv, 1.0f);
      acc += fmap[((long)b * HW_ + pix) * C_ + c] * inv;
    }
  }
  out[(((long)b * C_ + c) * OH_ + oh) * OW_ + ow] = acc * (1.0f / 16.0f);
}

extern "C" void kernel_launch(void* const* d_in, const int* in_sizes, int n_in,
                              void* d_out, int out_size, void* d_ws, size_t ws_size,
                              hipStream_t stream) {
  const float* xyz_feats = (const float*)d_in[0];  // (B,C,M)
  const float* points    = (const float*)d_in[1];  // (B,N,3)
  const float* centers   = (const float*)d_in[2];  // (B,M,3)
  const float* Kmat      = (const float*)d_in[3];  // 3x3

  float* ws = (float*)d_ws;
  const size_t fmap_n = (size_t)B_ * HW_ * C_;     // 51,380,224 floats
  const size_t cnt_n  = (size_t)B_ * HW_;          // 200,704
  const size_t ft_n   = (size_t)B_ * M_ * C_;      // 1,048,576
  float* fmap = ws;
  float* cnt  = fmap + fmap_n;
  float* Ft   = cnt + cnt_n;
  float* rec  = Ft + ft_n;                         // B*N*8 floats

  // Zero the accumulation buffers every call (workspace is not re-poisoned).
  hipMemsetAsync(fmap, 0, (fmap_n + cnt_n) * sizeof(float), stream);

  transpose_feats_kernel<<<B_ * M_, 256, 0, stream>>>(xyz_feats, Ft);
  topk_wmma_kernel<<<(B_ * N_) / 128, 256, 0, stream>>>(points, centers, Kmat, rec);
  scatter_kernel<<<B_ * N_, 256, 0, stream>>>(Ft, rec, fmap, cnt);
  pool_kernel<<<B_ * OH_ * OW_, 256, 0, stream>>>(fmap, cnt, (float*)d_out);
}